// GNN_TLearner_40037685133338
// MI455X (gfx1250) — compile-verified
//
#include <hip/hip_runtime.h>
#include <math.h>

typedef float v2f __attribute__((ext_vector_type(2)));
typedef float v8f __attribute__((ext_vector_type(8)));

// D = A(16x4 f32) * B(4x16 f32) + C(16x16 f32), wave32, full f32 precision.
#define WMMA_F32X4(a, b, c) \
  __builtin_amdgcn_wmma_f32_16x16x4_f32(false, (a), false, (b), (short)0, (c), false, false)

#define NODE_DIM 64
#define E_HID 16
#define EGO_HID 32
#define PEER_HID 48
#define HDIM 112
#define F_HID 64

// ---------------- init ----------------
__global__ void k_zero(float* __restrict__ p, int n) {
  int i = blockIdx.x * blockDim.x + threadIdx.x;
  if (i < n) p[i] = 0.0f;
}

// Wcat[64][80] = [ W_ego(64x32) | W_peer_x(64x48) ]
__global__ void k_build_wcat(const float* __restrict__ W_ego, const float* __restrict__ W_peer,
                             float* __restrict__ Wcat) {
  int i = blockIdx.x * blockDim.x + threadIdx.x;
  if (i >= 64 * 80) return;
  int k = i / 80, j = i % 80;
  Wcat[i] = (j < EGO_HID) ? W_ego[k * EGO_HID + j] : W_peer[k * PEER_HID + (j - EGO_HID)];
}

// ---------------- edge pass 1: deg, esum, edge-MLP scatter ----------------
__global__ void k_edge_pass1(const float* __restrict__ ea, const int* __restrict__ dst,
                             const float* __restrict__ W_edge, const float* __restrict__ b_edge,
                             float* __restrict__ deg, float* __restrict__ esum,
                             float* __restrict__ hagg, int E) {
  int e = blockIdx.x * blockDim.x + threadIdx.x;
  if (e >= E) return;
  float a0 = ea[2 * e + 0];
  float a1 = ea[2 * e + 1];
  int d = dst[e];
  atomicAdd(&deg[d], 1.0f);
  atomicAdd(&esum[2 * d + 0], a0);
  atomicAdd(&esum[2 * d + 1], a1);
  float* hd = hagg + (size_t)E_HID * d;
#pragma unroll
  for (int j = 0; j < E_HID; ++j) {
    float h = fmaxf(a0 * W_edge[j] + a1 * W_edge[E_HID + j] + b_edge[j], 0.0f);
    atomicAdd(&hd[j], h);
  }
}

// ---------------- edge pass 2: 2-hop gather/scatter (deg_inv applied later) ----------------
__global__ void k_edge_pass2(const int* __restrict__ src, const int* __restrict__ dst,
                             const float* __restrict__ hagg, float* __restrict__ h2, int E) {
  long t = (long)blockIdx.x * blockDim.x + threadIdx.x;
  int e = (int)(t >> 2);
  if (e >= E) return;
  int q = (int)(t & 3);
  int s = src[e], d = dst[e];
  const float4 v = *(const float4*)(hagg + (size_t)E_HID * s + 4 * q);
  float* p = h2 + (size_t)E_HID * d + 4 * q;
  atomicAdd(p + 0, v.x);
  atomicAdd(p + 1, v.y);
  atomicAdd(p + 2, v.z);
  atomicAdd(p + 3, v.w);
}

// ---------------- WMMA GEMM: x[N,64] @ Wcat[64,80] -> femb cols {0..31, 64..111} (pre-bias) ----
// blockDim = 160 (5 waves); wave w computes a 16x16 tile, blockIdx = row block of 16.
__global__ void k_gemm_ego_peer(const float* __restrict__ x, const float* __restrict__ Wcat,
                                float* __restrict__ femb) {
  int lane = threadIdx.x & 31;
  int w = threadIdx.x >> 5;  // 0..4 -> column tile of Wcat
  int row0 = blockIdx.x * 16;
  int m = lane & 15;   // M (and N) index within tile
  int kh = lane >> 4;  // K half: lanes 0-15 hold K%4 in {0,1}; lanes 16-31 hold {2,3}
  const float* arow = x + (size_t)(row0 + m) * NODE_DIM + 2 * kh;
  const float* bcol = Wcat + w * 16 + m;
  v8f acc = {};
#pragma unroll
  for (int k0 = 0; k0 < NODE_DIM; k0 += 4) {
    v2f a = *(const v2f*)(arow + k0);  // A[m, k0+2*kh .. +1]
    int kb = k0 + 2 * kh;
    v2f b;
    b[0] = bcol[kb * 80];        // B[kb,   n]
    b[1] = bcol[(kb + 1) * 80];  // B[kb+1, n]
    acc = WMMA_F32X4(a, b, acc);
  }
  // map Wcat col tile -> f_emb column: tiles 0,1 -> h_ego cols 0..31; tiles 2..4 -> peer cols 64..111
  int outc = ((w < 2) ? (w * 16) : (64 + (w - 2) * 16)) + m;
  int rbase = row0 + 8 * kh;
#pragma unroll
  for (int v = 0; v < 8; ++v) femb[(size_t)(rbase + v) * HDIM + outc] = acc[v];
}

// ---------------- per-node assembly of f_emb (biases, deg scaling, relu) ----------------
__global__ void k_node_assemble(const float* __restrict__ deg, const float* __restrict__ esum,
                                const float* __restrict__ hagg, const float* __restrict__ h2,
                                const float* __restrict__ b_ego, const float* __restrict__ b_peer,
                                const float* __restrict__ W_peer, float* __restrict__ femb) {
  int n = blockIdx.x;
  int c = threadIdx.x;
  if (c >= HDIM) return;
  float dg = deg[n];
  float dinv = (dg > 0.0f) ? rsqrtf(dg) : 0.0f;
  float* row = femb + (size_t)n * HDIM;
  float v;
  if (c < 32) {  // h_ego
    v = fmaxf(row[c] + b_ego[c], 0.0f);
  } else if (c < 48) {  // h_edge_agg
    v = fmaxf(hagg[(size_t)E_HID * n + (c - 32)], 0.0f);
  } else if (c < 64) {  // h_edge_2hop = deg_inv * accumulated gather
    v = fmaxf(dinv * h2[(size_t)E_HID * n + (c - 48)], 0.0f);
  } else {  // h_peer, factored: relu(dinv*(deg*(x@Wp_x + b) + esum@Wp_e))
    int j = c - 64;
    float pre = row[c] + b_peer[j];
    float s = dg * pre + esum[2 * n + 0] * W_peer[64 * PEER_HID + j]
                       + esum[2 * n + 1] * W_peer[65 * PEER_HID + j];
    v = fmaxf(dinv * s, 0.0f);
  }
  row[c] = v;
}

// ---------------- WMMA GEMM: femb[N,112] @ W_emb[112,64] + fused bias/relu/BN/tanh ----------
// blockDim = 128 (4 waves); wave w = 16-col tile; blockIdx = row block of 16.
__global__ void k_gemm_emb(const float* __restrict__ femb, const float* __restrict__ W_emb,
                           const float* __restrict__ b_emb, const float* __restrict__ gamma,
                           const float* __restrict__ beta, const float* __restrict__ mean,
                           const float* __restrict__ var, float* __restrict__ emb_out) {
  int lane = threadIdx.x & 31;
  int w = threadIdx.x >> 5;  // 0..3
  int row0 = blockIdx.x * 16;
  int m = lane & 15;
  int kh = lane >> 4;
  const float* arow = femb + (size_t)(row0 + m) * HDIM + 2 * kh;
  const float* bcol = W_emb + w * 16 + m;
  v8f acc = {};
#pragma unroll
  for (int k0 = 0; k0 < HDIM; k0 += 4) {
    v2f a = *(const v2f*)(arow + k0);
    int kb = k0 + 2 * kh;
    v2f b;
    b[0] = bcol[kb * F_HID];
    b[1] = bcol[(kb + 1) * F_HID];
    acc = WMMA_F32X4(a, b, acc);
  }
  int col = w * 16 + m;
  float bb = b_emb[col];
  float sc = rsqrtf(var[col] + 1e-5f) * gamma[col];
  float mn = mean[col], bt = beta[col];
#pragma unroll
  for (int v = 0; v < 8; ++v) {
    int row = row0 + 8 * kh + v;
    float t = fmaxf(acc[v] + bb, 0.0f);
    t = (t - mn) * sc + bt;
    emb_out[(size_t)row * F_HID + col] = tanhf(t);
  }
}

// ---------------- heads: y0/y1 = emb @ W{0,1} + b ----------------
__global__ void k_heads(const float* __restrict__ emb, const float* __restrict__ W0,
                        const float* __restrict__ b0, const float* __restrict__ W1,
                        const float* __restrict__ b1, float* __restrict__ y0,
                        float* __restrict__ y1, int N) {
  int n = blockIdx.x * blockDim.x + threadIdx.x;
  if (n >= N) return;
  const float* r = emb + (size_t)n * F_HID;
  float s0 = 0.0f, s1 = 0.0f;
#pragma unroll
  for (int c = 0; c < F_HID; ++c) {
    float e = r[c];
    s0 += e * W0[c];
    s1 += e * W1[c];
  }
  y0[n] = s0 + b0[0];
  y1[n] = s1 + b1[0];
}

extern "C" void kernel_launch(void* const* d_in, const int* in_sizes, int n_in,
                              void* d_out, int out_size, void* d_ws, size_t ws_size,
                              hipStream_t stream) {
  const float* x          = (const float*)d_in[0];
  const float* edge_attrs = (const float*)d_in[1];
  // d_in[2] = treatments (unused in forward)
  const int*   src        = (const int*)d_in[3];
  const int*   dst        = (const int*)d_in[4];
  const float* W_peer     = (const float*)d_in[5];
  const float* b_peer     = (const float*)d_in[6];
  const float* W_ego      = (const float*)d_in[7];
  const float* b_ego      = (const float*)d_in[8];
  const float* W_edge     = (const float*)d_in[9];
  const float* b_edge     = (const float*)d_in[10];
  const float* W_emb      = (const float*)d_in[11];
  const float* b_emb      = (const float*)d_in[12];
  const float* bn_gamma   = (const float*)d_in[13];
  const float* bn_beta    = (const float*)d_in[14];
  const float* bn_mean    = (const float*)d_in[15];
  const float* bn_var     = (const float*)d_in[16];
  const float* W0         = (const float*)d_in[17];
  const float* b0         = (const float*)d_in[18];
  const float* W1         = (const float*)d_in[19];
  const float* b1         = (const float*)d_in[20];

  const int N = in_sizes[0] / NODE_DIM;  // 50000 (multiple of 16)
  const int E = in_sizes[3];             // 1,600,000

  // workspace layout (floats): deg[N] | esum[2N] | hagg[16N] | h2[16N] | femb[112N] | Wcat[64*80]
  float* ws   = (float*)d_ws;
  float* deg  = ws;
  float* esum = ws + (size_t)N;
  float* hagg = ws + (size_t)3 * N;
  float* h2   = ws + (size_t)19 * N;
  float* femb = ws + (size_t)35 * N;
  float* Wcat = ws + (size_t)147 * N;

  float* y0  = (float*)d_out;
  float* y1  = y0 + N;
  float* emb = y0 + (size_t)2 * N;

  const int zn = 35 * N;  // accumulators only
  k_zero<<<(zn + 255) / 256, 256, 0, stream>>>(ws, zn);
  k_build_wcat<<<(64 * 80 + 255) / 256, 256, 0, stream>>>(W_ego, W_peer, Wcat);
  k_edge_pass1<<<(E + 255) / 256, 256, 0, stream>>>(edge_attrs, dst, W_edge, b_edge,
                                                    deg, esum, hagg, E);
  k_edge_pass2<<<(4 * E + 255) / 256, 256, 0, stream>>>(src, dst, hagg, h2, E);
  k_gemm_ego_peer<<<N / 16, 160, 0, stream>>>(x, Wcat, femb);
  k_node_assemble<<<N, 128, 0, stream>>>(deg, esum, hagg, h2, b_ego, b_peer, W_peer, femb);
  k_gemm_emb<<<N / 16, 128, 0, stream>>>(femb, W_emb, b_emb, bn_gamma, bn_beta,
                                         bn_mean, bn_var, emb);
  k_heads<<<(N + 255) / 256, 256, 0, stream>>>(emb, W0, b0, W1, b1, y0, y1, N);
}